// GNN_26371099197447
// MI455X (gfx1250) — compile-verified
//
#include <hip/hip_runtime.h>
#include <math.h>

#define N_NODES 10000
#define E_EDGES 320000
#define DIN     256
#define HID     128
#define M_TRAIN 100000
#define P_PAIRS 15000
#define T_C     0.1f
#define TAU0_R  0.5f
#define LAMBDA1_C 1.0f
#define THRESH_C  0.95f

#define NTILES  625          /* N_NODES / 16 */
#define BSTRIDE 260          /* padded LDS row stride (floats): 260 % 64 == 4 -> conflict-free */

typedef __attribute__((ext_vector_type(2))) float v2f;
typedef __attribute__((ext_vector_type(8))) float v8f;

__device__ __forceinline__ float wave_sum(float v) {
    #pragma unroll
    for (int o = 16; o > 0; o >>= 1) v += __shfl_xor(v, o, 32);
    return v;
}

__device__ __forceinline__ void atomic_add_f32(float* p, float v) {
    __hip_atomic_fetch_add(p, v, __ATOMIC_RELAXED, __HIP_MEMORY_SCOPE_AGENT);
}

// ---------------- GEMM: sup = x @ W + b   (N x DIN) @ (DIN x HID) --------------
// One wave per 16x16 output tile, K-loop of 4 via V_WMMA_F32_16X16X4_F32.
__global__ __launch_bounds__(256) void gemm_xw_kernel(
    const float* __restrict__ x0, const float* __restrict__ W0, const float* __restrict__ b0,
    const float* __restrict__ x1, const float* __restrict__ W1, const float* __restrict__ b1,
    float* __restrict__ sup0, float* __restrict__ sup1)
{
    const int lane = threadIdx.x & 31;
    const int wave = threadIdx.x >> 5;
    const float* x; const float* W; const float* b; float* sup;
    if (blockIdx.z == 0) { x = x0; W = W0; b = b0; sup = sup0; }
    else                 { x = x1; W = W1; b = b1; sup = sup1; }

    const int i0 = blockIdx.x * 16;
    const int j0 = wave * 16;          // 8 waves -> HID/16 = 8 tiles
    const int lo = lane & 15;
    const int hi = lane >> 4;

    const float* arow = x + (long)(i0 + lo) * DIN + 2 * hi;
    v8f c = {};
    for (int k0 = 0; k0 < DIN; k0 += 4) {
        v2f a = *(const v2f*)(arow + k0);
        const int kb = k0 + 2 * hi;
        v2f bf;
        bf.x = W[(long)kb * HID + j0 + lo];
        bf.y = W[(long)(kb + 1) * HID + j0 + lo];
        c = __builtin_amdgcn_wmma_f32_16x16x4_f32(false, a, false, bf, (short)0, c,
                                                  false, false);
    }
    const float bias = b[j0 + lo];
    #pragma unroll
    for (int r = 0; r < 8; ++r) {
        const int m = i0 + r + 8 * hi;
        sup[(long)m * HID + j0 + lo] = c[r] + bias;
    }
}

// ---------------- Edge scatter: h[dst] += sup[src] * val -----------------------
__global__ __launch_bounds__(256) void scatter_kernel(
    const int* __restrict__ src0, const int* __restrict__ dst0, const float* __restrict__ val0,
    const int* __restrict__ src1, const int* __restrict__ dst1, const float* __restrict__ val1,
    const float* __restrict__ sup0, const float* __restrict__ sup1,
    float* __restrict__ h0, float* __restrict__ h1)
{
    const int lane = threadIdx.x & 31;
    const int wave = threadIdx.x >> 5;
    const long e = (long)blockIdx.x * 8 + wave;
    if (e >= E_EDGES) return;
    const int* src; const int* dst; const float* val; const float* sup; float* h;
    if (blockIdx.y == 0) { src = src0; dst = dst0; val = val0; sup = sup0; h = h0; }
    else                 { src = src1; dst = dst1; val = val1; sup = sup1; h = h1; }
    const int   s = src[e];
    const int   d = dst[e];
    const float v = val[e];
    const float* sp = sup + (long)s * HID;
    float*       hp = h   + (long)d * HID;
    #pragma unroll
    for (int c = 0; c < 4; ++c)
        atomic_add_f32(hp + c * 32 + lane, sp[c * 32 + lane] * v);
}

// ---------------- Row normalize: x_all = [h0n | h1n], xn = x_all/sqrt(2) -------
__global__ __launch_bounds__(256) void rownorm_kernel(
    const float* __restrict__ h0, const float* __restrict__ h1,
    float* __restrict__ xall, float* __restrict__ xn)
{
    const int lane = threadIdx.x & 31;
    const int wave = threadIdx.x >> 5;
    const int row = blockIdx.x * 8 + wave;        // 1250*8 = 10000 exact
    const float* p0 = h0 + (long)row * HID;
    const float* p1 = h1 + (long)row * HID;
    float v0[4], v1[4], s0 = 0.f, s1 = 0.f;
    #pragma unroll
    for (int c = 0; c < 4; ++c) {
        v0[c] = p0[c * 32 + lane]; s0 += v0[c] * v0[c];
        v1[c] = p1[c * 32 + lane]; s1 += v1[c] * v1[c];
    }
    s0 = wave_sum(s0);
    s1 = wave_sum(s1);
    const float i0 = rsqrtf(s0);
    const float i1 = rsqrtf(s1);
    float* xa = xall + (long)row * 256;
    float* xr = xn   + (long)row * 256;
    const float rn = 0.70710678118654752f;        // each half unit-norm -> row norm sqrt(2)
    #pragma unroll
    for (int c = 0; c < 4; ++c) {
        const float a = v0[c] * i0;
        const float b = v1[c] * i1;
        xa[c * 32 + lane]       = a;  xr[c * 32 + lane]       = a * rn;
        xa[128 + c * 32 + lane] = b;  xr[128 + c * 32 + lane] = b * rn;
    }
}

// ---------------- neg_s[i] = dot(xn[i], xn[neg_row[i]]) ------------------------
__global__ __launch_bounds__(256) void negs_kernel(
    const float* __restrict__ xn, const int* __restrict__ neg_row, float* __restrict__ negs)
{
    const int lane = threadIdx.x & 31;
    const int wave = threadIdx.x >> 5;
    const int row = blockIdx.x * 8 + wave;
    const int nr  = neg_row[row];
    const float* a = xn + (long)row * 256;
    const float* b = xn + (long)nr  * 256;
    float d = 0.f;
    #pragma unroll
    for (int c = 0; c < 8; ++c) d += a[c * 32 + lane] * b[c * 32 + lane];
    d = wave_sum(d);
    if (lane == 0) negs[row] = d;
}

// ---------------- lc loss over P pairs -----------------------------------------
__global__ __launch_bounds__(256) void lc_kernel(
    const float* __restrict__ xall, const int* __restrict__ na, const int* __restrict__ nb,
    const int* __restrict__ nl, float* __restrict__ acc)
{
    const int lane = threadIdx.x & 31;
    const int wave = threadIdx.x >> 5;
    const int p = blockIdx.x * 8 + wave;          // 1875*8 = 15000 exact
    const float* A = xall + (long)na[p] * 256;
    const float* B = xall + (long)nb[p] * 256;
    float dot = 0.f, sa = 0.f, sb = 0.f;
    #pragma unroll
    for (int c = 0; c < 8; ++c) {
        const float av = A[c * 32 + lane];
        const float bv = B[c * 32 + lane];
        dot += av * bv; sa += av * av; sb += bv * bv;
    }
    dot = wave_sum(dot); sa = wave_sum(sa); sb = wave_sum(sb);
    if (lane == 0) {
        const float sim = dot * rsqrtf(sa) * rsqrtf(sb);
        const float z = sim / TAU0_R;
        const float lsig = (z < 0.f) ? (z - log1pf(expf(z))) : (-log1pf(expf(-z)));
        const float g = ldexpf(1.0f, -(nl[p] + 1));   // 0.5^(L+1)
        atomic_add_f32(acc, g * lsig);
    }
}

// ---------------- loss_real over M triples -------------------------------------
__global__ __launch_bounds__(256) void real_kernel(
    const float* __restrict__ xall, const int* __restrict__ trainset,
    const int* __restrict__ negi, float* __restrict__ acc)
{
    const int lane = threadIdx.x & 31;
    const int wave = threadIdx.x >> 5;
    const int m = blockIdx.x * 8 + wave;          // 12500*8 = 100000 exact
    const float* R = xall + (long)trainset[2 * m]     * 256;
    const float* E = xall + (long)trainset[2 * m + 1] * 256;
    const float* G = xall + (long)negi[m]             * 256;
    float rr = 0.f, ee = 0.f, gg = 0.f, re = 0.f, rg = 0.f;
    #pragma unroll
    for (int c = 0; c < 8; ++c) {
        const float rv = R[c * 32 + lane];
        const float ev = E[c * 32 + lane];
        const float gv = G[c * 32 + lane];
        rr += rv * rv; ee += ev * ev; gg += gv * gv; re += rv * ev; rg += rv * gv;
    }
    rr = wave_sum(rr); ee = wave_sum(ee); gg = wave_sum(gg);
    re = wave_sum(re); rg = wave_sum(rg);
    if (lane == 0) {
        const float ir = rsqrtf(rr);
        const float pos = re * ir * rsqrtf(ee);
        const float neg = rg * ir * rsqrtf(gg);
        const float t = (pos - T_C) * (1.0f / (1.0f - T_C));
        atomic_add_f32(acc, t * t * log1pf(expf(neg - pos)));
    }
}

// ---------------- Fused S = xn @ xn^T + pseudo loss (never materialize S) ------
// Block: 8 i-tiles (one per wave) x one 64-column j-panel staged in LDS.
// B panel (64 rows x 256 K) is shared by all 8 waves -> stage once via b128
// loads, then feed WMMA B-fragments with bank-conflict-free ds_load_b64.
__global__ __launch_bounds__(256) void pseudo_kernel(
    const float* __restrict__ xn, const float* __restrict__ negs, float* __restrict__ acc)
{
    __shared__ float smem[64 * BSTRIDE];          // ~65 KB of the 320 KB WGP LDS

    const int tid  = threadIdx.x;
    const int lane = tid & 31;
    const int wave = tid >> 5;
    const int bi0      = blockIdx.x * 8;          // first i-tile of this block
    const int jt_base  = blockIdx.y * 4;          // first j-tile of the panel
    if (jt_base + 3 < bi0) return;                // whole block below diagonal (uniform)

    // ---- stage B panel: xn rows [jt_base*16, jt_base*16 + 64) ----
    const int jrow0 = jt_base * 16;
    #pragma unroll
    for (int q = 0; q < 16; ++q) {
        const int idx = q * 256 + tid;            // 0..4095 float4 slots
        const int row = idx >> 6;                 // 64 float4 per row
        const int c4  = idx & 63;
        if (jrow0 + row < N_NODES) {
            const float4 v = *(const float4*)(xn + (long)(jrow0 + row) * 256 + c4 * 4);
            *(float4*)(&smem[row * BSTRIDE + c4 * 4]) = v;
        }
    }
    __syncthreads();

    const int bi = bi0 + wave;
    const int lo = lane & 15;
    const int hi = lane >> 4;
    const int i0 = bi * 16;
    const bool wave_active = (bi < NTILES) && (jt_base + 3 >= bi);

    float lacc = 0.f;
    if (wave_active) {                            // wave-uniform: EXEC stays all-1s
        bool valid[4];
        #pragma unroll
        for (int jt = 0; jt < 4; ++jt) {
            const int bjt = jt_base + jt;
            valid[jt] = (bjt >= bi) && (bjt < NTILES);
        }
        const float* arow  = xn + (long)(i0 + lo) * 256 + 2 * hi;
        const float* bbase = &smem[lo * BSTRIDE + 2 * hi];

        v8f c[4] = {};
        for (int k0 = 0; k0 < 256; k0 += 4) {
            const v2f a = *(const v2f*)(arow + k0);
            #pragma unroll
            for (int jt = 0; jt < 4; ++jt) {
                if (valid[jt]) {
                    const v2f b = *(const v2f*)(bbase + jt * 16 * BSTRIDE + k0);
                    c[jt] = __builtin_amdgcn_wmma_f32_16x16x4_f32(false, a, false, b,
                                                                  (short)0, c[jt],
                                                                  false, false);
                }
            }
        }

        #pragma unroll
        for (int jt = 0; jt < 4; ++jt) {
            if (!valid[jt]) continue;
            const int n = (jt_base + jt) * 16 + lo;
            #pragma unroll
            for (int r = 0; r < 8; ++r) {
                const int   m = i0 + r + 8 * hi;
                const float s = c[jt][r];
                if (n > m && s > THRESH_C) {
                    const float t = (s - T_C) * (1.0f / (1.0f - T_C));
                    lacc += t * t * log1pf(expf(negs[m] - s));
                }
            }
        }
    }
    lacc = wave_sum(lacc);
    if (wave_active && lane == 0) atomic_add_f32(acc, lacc);
}

// ---------------- Final combine ------------------------------------------------
__global__ void finalize_kernel(const float* __restrict__ acc, float* __restrict__ out)
{
    const float lc = -acc[0] / (float)P_PAIRS;
    out[0] = acc[1] + acc[2] + LAMBDA1_C * lc;
}

extern "C" void kernel_launch(void* const* d_in, const int* in_sizes, int n_in,
                              void* d_out, int out_size, void* d_ws, size_t ws_size,
                              hipStream_t stream)
{
    const float* x0   = (const float*)d_in[0];
    const float* x1   = (const float*)d_in[1];
    const int*   a0s  = (const int*)d_in[2];
    const int*   a0d  = (const int*)d_in[3];
    const float* a0v  = (const float*)d_in[4];
    const int*   a1s  = (const int*)d_in[5];
    const int*   a1d  = (const int*)d_in[6];
    const float* a1v  = (const float*)d_in[7];
    const int*   trn  = (const int*)d_in[8];
    const int*   ngi  = (const int*)d_in[9];
    const int*   ngr  = (const int*)d_in[10];
    const int*   na   = (const int*)d_in[11];
    const int*   nb   = (const int*)d_in[12];
    const int*   nl   = (const int*)d_in[13];
    const float* W0   = (const float*)d_in[14];
    const float* b0   = (const float*)d_in[15];
    const float* W1   = (const float*)d_in[16];
    const float* b1   = (const float*)d_in[17];

    float* xall = (float*)d_out;                  // first N*256 floats of output
    float* loss = xall + (long)N_NODES * 256;

    float* ws   = (float*)d_ws;
    float* sup0 = ws;                             // N*128
    float* sup1 = sup0 + (long)N_NODES * HID;     // N*128
    float* h0   = sup1 + (long)N_NODES * HID;     // N*128 (zeroed)
    float* h1   = h0   + (long)N_NODES * HID;     // N*128 (zeroed)
    float* xn   = h1   + (long)N_NODES * HID;     // N*256
    float* negs = xn   + (long)N_NODES * 256;     // N
    float* acc  = negs + N_NODES;                 // [lc_sum, real, pseudo, pad]

    hipMemsetAsync(h0,  0, 2 * (size_t)N_NODES * HID * sizeof(float), stream);
    hipMemsetAsync(acc, 0, 4 * sizeof(float), stream);

    gemm_xw_kernel<<<dim3(N_NODES / 16, 1, 2), 256, 0, stream>>>(
        x0, W0, b0, x1, W1, b1, sup0, sup1);

    scatter_kernel<<<dim3(E_EDGES / 8, 2), 256, 0, stream>>>(
        a0s, a0d, a0v, a1s, a1d, a1v, sup0, sup1, h0, h1);

    rownorm_kernel<<<N_NODES / 8, 256, 0, stream>>>(h0, h1, xall, xn);

    negs_kernel<<<N_NODES / 8, 256, 0, stream>>>(xn, ngr, negs);

    lc_kernel<<<P_PAIRS / 8, 256, 0, stream>>>(xall, na, nb, nl, acc + 0);

    real_kernel<<<M_TRAIN / 8, 256, 0, stream>>>(xall, trn, ngi, acc + 1);

    pseudo_kernel<<<dim3(79, 157), 256, 0, stream>>>(xn, negs, acc + 2);

    finalize_kernel<<<1, 1, 0, stream>>>(acc, loss);
}